// FreeNetTrig_2319282340034
// MI455X (gfx1250) — compile-verified
//
#include <hip/hip_runtime.h>

typedef __attribute__((ext_vector_type(2))) float v2f;
typedef __attribute__((ext_vector_type(4))) float v4f;
typedef __attribute__((ext_vector_type(8))) float v8f;

#define B_TOT 2048
#define IND   512
#define HD    256
#define OUTD  256
#define BT    32          // batch rows per workgroup
#define NW    8           // waves per block (256 threads, wave32)
#define XT_STRIDE 33      // LDS row stride for xT  (odd -> conflict-free)
#define H_STRIDE  33      // LDS row stride for hT

// D = A(16x4 f32) * B(4x16 f32) + C(16x16 f32), wave32 WMMA.
__device__ __forceinline__ v8f wmma_f32(v2f a, v2f b, v8f c) {
    return __builtin_amdgcn_wmma_f32_16x16x4_f32(
        /*neg_a=*/false, a, /*neg_b=*/false, b,
        /*c_mod=*/(short)0, c, /*reuse_a=*/false, /*reuse_b=*/false);
}

__global__ __launch_bounds__(256)
void freenet_fused_kernel(const float* __restrict__ x,      // [2048,512]
                          const float* __restrict__ W_in,   // [256,1024]
                          const float* __restrict__ b_in,   // [256]
                          const float* __restrict__ W_hh,   // [255,256,256]
                          const float* __restrict__ b_hh,   // [255,256]
                          const float* __restrict__ W_out,  // [256,256]
                          const float* __restrict__ b_out,  // [256]
                          float* __restrict__ out)          // [2048,256]
{
    extern __shared__ float smem[];
    float* xT = smem;                       // [512][XT_STRIDE]  raw x, transposed
    float* hT = smem + IND * XT_STRIDE;     // [256][H_STRIDE]   h,    transposed

    const int tid    = threadIdx.x;
    const int wave   = tid >> 5;
    const int lane   = tid & 31;
    const int lane16 = lane & 15;
    const int laneHi = lane >> 4;           // 0: K=k0,k0+1  1: K=k0+2,k0+3
    const int koffA  = laneHi * 2;
    const int b0     = blockIdx.x * BT;

    const v8f zero8 = {0.f,0.f,0.f,0.f,0.f,0.f,0.f,0.f};

    // ---------------- Stage x tile -> LDS (transposed), coalesced float4 ----
    // BT*IND = 16384 floats = 4096 float4 -> 16 iters over 256 threads
    for (int it = 0; it < (BT * IND) / (256 * 4); ++it) {
        int idx4 = it * 256 + tid;
        int k4   = (idx4 & (IND / 4 - 1)) * 4;   // 0..508
        int b    = idx4 >> 7;                    // 0..31
        v4f v = *(const v4f*)(x + (size_t)(b0 + b) * IND + k4);
        xT[(k4 + 0) * XT_STRIDE + b] = v.x;
        xT[(k4 + 1) * XT_STRIDE + b] = v.y;
        xT[(k4 + 2) * XT_STRIDE + b] = v.z;
        xT[(k4 + 3) * XT_STRIDE + b] = v.w;
    }
    __syncthreads();

    // ---------------- Phase 0: h = sqrelu([sin x, cos x] @ W_in^T + b_in) --
    // Split K-loop: k < 512 uses sin(x[k]), k >= 512 uses cos(x[k-512]).
    for (int t = wave; t < HD / 16; t += NW) {
        v8f acc[2] = {zero8, zero8};
        const float* wrow = W_in + (size_t)(t * 16 + lane16) * (2 * IND);

        #pragma unroll 4
        for (int kc = 0; kc < IND / 4; ++kc) {          // sin half
            int ka = kc * 4 + koffA;
            v2f bf = *(const v2f*)(wrow + ka);
            v2f a0, a1;
            a0.x = __sinf(xT[ka * H_STRIDE + lane16]);
            a0.y = __sinf(xT[(ka + 1) * H_STRIDE + lane16]);
            a1.x = __sinf(xT[ka * H_STRIDE + 16 + lane16]);
            a1.y = __sinf(xT[(ka + 1) * H_STRIDE + 16 + lane16]);
            acc[0] = wmma_f32(a0, bf, acc[0]);
            acc[1] = wmma_f32(a1, bf, acc[1]);
        }
        #pragma unroll 4
        for (int kc = 0; kc < IND / 4; ++kc) {          // cos half
            int ka = kc * 4 + koffA;
            v2f bf = *(const v2f*)(wrow + IND + ka);
            v2f a0, a1;
            a0.x = __cosf(xT[ka * H_STRIDE + lane16]);
            a0.y = __cosf(xT[(ka + 1) * H_STRIDE + lane16]);
            a1.x = __cosf(xT[ka * H_STRIDE + 16 + lane16]);
            a1.y = __cosf(xT[(ka + 1) * H_STRIDE + 16 + lane16]);
            acc[0] = wmma_f32(a0, bf, acc[0]);
            acc[1] = wmma_f32(a1, bf, acc[1]);
        }

        int j = t * 16 + lane16;
        float bias = b_in[j];
        #pragma unroll
        for (int m = 0; m < 2; ++m)
            #pragma unroll
            for (int r = 0; r < 8; ++r) {
                int bb = m * 16 + laneHi * 8 + r;
                float z  = acc[m][r] + bias;
                float rl = fmaxf(z, 0.f);
                hT[j * H_STRIDE + bb] = rl * rl;
            }
    }
    __syncthreads();

    // ---------------- Scan: 255 triangular masked steps --------------------
    for (int i = 0; i < HD - 1; ++i) {
        const float* Wi = W_hh + (size_t)i * HD * HD;
        const float* bi = b_hh + (size_t)i * HD;

        // Prefetch next step's weights (L2-resident, warms WGP path)
        if (i + 1 < HD - 1) {
            const float* Wn = W_hh + (size_t)(i + 1) * HD * HD;
            #pragma unroll
            for (int q = 0; q < 8; ++q)
                __builtin_prefetch(Wn + ((tid << 3) + q) * 32, 0, 1);
        }

        const int t0   = (i + 1) >> 4;   // first N-tile containing j > i
        const int kcf  = (i + 1) >> 2;   // fully valid K-chunks (k..k+3 <= i)
        const int krem = (i + 1) & 3;

        for (int t = t0 + wave; t < HD / 16; t += NW) {
            v8f acc[2] = {zero8, zero8};
            const float* wrow = Wi + (size_t)(t * 16 + lane16) * HD;
            #pragma unroll 4
            for (int kc = 0; kc < kcf; ++kc) {
                int ka = kc * 4 + koffA;
                v2f bf = *(const v2f*)(wrow + ka);
                v2f a0, a1;
                a0.x = hT[ka * H_STRIDE + lane16];
                a0.y = hT[(ka + 1) * H_STRIDE + lane16];
                a1.x = hT[ka * H_STRIDE + 16 + lane16];
                a1.y = hT[(ka + 1) * H_STRIDE + 16 + lane16];
                acc[0] = wmma_f32(a0, bf, acc[0]);
                acc[1] = wmma_f32(a1, bf, acc[1]);
            }
            if (krem) {                  // boundary chunk: zero B where k > i
                int ka = kcf * 4 + koffA;
                v2f bf;
                bf.x = (ka     <= i) ? wrow[ka]     : 0.f;
                bf.y = (ka + 1 <= i) ? wrow[ka + 1] : 0.f;
                v2f a0, a1;
                a0.x = hT[ka * H_STRIDE + lane16];
                a0.y = hT[(ka + 1) * H_STRIDE + lane16];
                a1.x = hT[ka * H_STRIDE + 16 + lane16];
                a1.y = hT[(ka + 1) * H_STRIDE + 16 + lane16];
                acc[0] = wmma_f32(a0, bf, acc[0]);
                acc[1] = wmma_f32(a1, bf, acc[1]);
            }
            // epilogue: h[:, j] += sqrelu(z)  only for j > i
            int j = t * 16 + lane16;
            if (j > i) {
                float bias = bi[j];
                #pragma unroll
                for (int m = 0; m < 2; ++m)
                    #pragma unroll
                    for (int r = 0; r < 8; ++r) {
                        int bb = m * 16 + laneHi * 8 + r;
                        float z  = acc[m][r] + bias;
                        float rl = fmaxf(z, 0.f);
                        hT[j * H_STRIDE + bb] += rl * rl;
                    }
            }
        }
        __syncthreads();   // writers (j>i) vs next step's readers (k<=i+1)
    }

    // ---------------- Output: out = h @ W_out^T + b_out --------------------
    for (int t = wave; t < OUTD / 16; t += NW) {
        v8f acc[2] = {zero8, zero8};
        const float* wrow = W_out + (size_t)(t * 16 + lane16) * HD;
        #pragma unroll 4
        for (int kc = 0; kc < HD / 4; ++kc) {
            int ka = kc * 4 + koffA;
            v2f bf = *(const v2f*)(wrow + ka);
            v2f a0, a1;
            a0.x = hT[ka * H_STRIDE + lane16];
            a0.y = hT[(ka + 1) * H_STRIDE + lane16];
            a1.x = hT[ka * H_STRIDE + 16 + lane16];
            a1.y = hT[(ka + 1) * H_STRIDE + 16 + lane16];
            acc[0] = wmma_f32(a0, bf, acc[0]);
            acc[1] = wmma_f32(a1, bf, acc[1]);
        }
        int j = t * 16 + lane16;
        float bias = b_out[j];
        #pragma unroll
        for (int m = 0; m < 2; ++m)
            #pragma unroll
            for (int r = 0; r < 8; ++r) {
                int bb = m * 16 + laneHi * 8 + r;          // batch row in tile
                out[(size_t)(b0 + bb) * OUTD + j] = acc[m][r] + bias;  // coalesced over lanes
            }
    }
}

extern "C" void kernel_launch(void* const* d_in, const int* in_sizes, int n_in,
                              void* d_out, int out_size, void* d_ws, size_t ws_size,
                              hipStream_t stream) {
    (void)in_sizes; (void)n_in; (void)out_size; (void)d_ws; (void)ws_size;
    const float* x     = (const float*)d_in[0];
    const float* W_in  = (const float*)d_in[1];
    const float* b_in  = (const float*)d_in[2];
    const float* W_hh  = (const float*)d_in[3];
    const float* b_hh  = (const float*)d_in[4];
    const float* W_out = (const float*)d_in[5];
    const float* b_out = (const float*)d_in[6];
    float* out = (float*)d_out;

    dim3 grid(B_TOT / BT);   // 64 workgroups
    dim3 block(256);         // 8 waves (wave32)
    size_t shmem = (size_t)(IND * XT_STRIDE + HD * H_STRIDE) * sizeof(float); // ~101 KB
    hipLaunchKernelGGL(freenet_fused_kernel, grid, block, shmem, stream,
                       x, W_in, b_in, W_hh, b_hh, W_out, b_out, out);
}